// FourierLayer_47450798686601
// MI455X (gfx1250) — compile-verified
//
#include <hip/hip_runtime.h>
#include <math.h>

// ---------------------------------------------------------------------------
// FNO layer on gfx1250:
//   x:(64,512,512) -> truncated DFT_W -> DFT_H -> per-mode complex mix (64->64)
//   -> iDFT_H -> fused [iDFT_W + skip GEMM + bias + gelu] -> out:(64,512,512)
// fp32 throughout; big GEMMs on V_WMMA_F32_16X16X4_F32.
// Mode-mix (the 134 MB HBM stream) is split 4-way over input channels for
// 4x more waves / outstanding loads (latency-hiding at 23.3 TB/s).
// ---------------------------------------------------------------------------

typedef float v8f __attribute__((ext_vector_type(8)));
typedef float v2f __attribute__((ext_vector_type(2)));

#define IN_C   64
#define OUT_C  64
#define HH     512
#define WW     512
#define MODES  64
#define HW     (HH * WW)          // 262144
#define TWO_PI 6.283185307179586f
#define ISPLIT 4                  // channel-split factor for the weight stream
#define ICHUNK (IN_C / ISPLIT)    // 16

// ---- WMMA f32 16x16x4 helpers ---------------------------------------------
// A (16x4): lanes 0-15 -> M=lane, VGPR0=K0,VGPR1=K1; lanes 16-31 -> K2,K3
// B (4x16): lanes 0-15 -> N=lane, VGPR0=K0-row,VGPR1=K1-row; lanes 16-31 -> K2,K3
// C/D (16x16): VGPR v, lanes 0-15 -> M=v, lanes 16-31 -> M=v+8; N=lane%16
__device__ __forceinline__ v8f wmma_f32(v2f a, v2f b, v8f c) {
    return __builtin_amdgcn_wmma_f32_16x16x4_f32(false, a, false, b, (short)0, c,
                                                 false, false);
}

__device__ __forceinline__ v2f load_a_tile(const float* __restrict__ p, int ld, int lane) {
    int m = lane & 15;
    int k = (lane >> 4) << 1;
    v2f a;
    a.x = p[(size_t)m * ld + k];
    a.y = p[(size_t)m * ld + k + 1];
    return a;
}

__device__ __forceinline__ v2f load_b_tile(const float* __restrict__ p, size_t ld, int lane) {
    int n = lane & 15;
    int k = (lane >> 4) << 1;
    v2f b;
    b.x = p[(size_t)k * ld + n];
    b.y = p[(size_t)(k + 1) * ld + n];
    return b;
}

// ---------------------------------------------------------------------------
// Kernel 0: twiddle tables (exact integer phase reduction: (a*b) & 511)
//   FW   (512 x 128): col 2*m2 = cos, col 2*m2+1 = -sin      (forward, e^{-i})
//   cosH/sinH (64 x 512): cos/sin(2*pi*m1*h/512)
//   DW   (128 x 512): row 2k = ck*cos/N^2, row 2k+1 = -ck*sin/N^2 (real iFFT)
// ---------------------------------------------------------------------------
__global__ void twiddle_kernel(float* __restrict__ FW, float* __restrict__ cosH,
                               float* __restrict__ sinH, float* __restrict__ DW) {
    int t = blockIdx.x * blockDim.x + threadIdx.x;
    if (t >= 32768) return;
    const float step = TWO_PI / 512.0f;
    {   // FW: t -> (w, m2)
        int w = t >> 6, m2 = t & 63;
        float th = (float)((w * m2) & 511) * step;
        FW[(size_t)w * 128 + 2 * m2]     = cosf(th);
        FW[(size_t)w * 128 + 2 * m2 + 1] = -sinf(th);
    }
    {   // cosH/sinH: t -> (m1, h)
        float th = (float)(((t >> 9) * (t & 511)) & 511) * step;
        cosH[t] = cosf(th);
        sinH[t] = sinf(th);
    }
    {   // DW: t -> (k, w); hermitian doubling + 1/(512*512)
        int k = t >> 9, w = t & 511;
        float th = (float)((k * w) & 511) * step;
        float s  = ((k == 0) ? 1.0f : 2.0f) / 262144.0f;
        DW[(size_t)(2 * k) * 512 + w]     = s * cosf(th);
        DW[(size_t)(2 * k + 1) * 512 + w] = -s * sinf(th);
    }
}

// ---------------------------------------------------------------------------
// Kernel 1 (WMMA): forward DFT along W, truncated to 64 modes.
//   Xw(32768 x 128) = x2d(32768 x 512) @ FW(512 x 128)
//   grid = 2048 row-tiles of 16, block = 128 (4 waves, 2 col-tiles each)
// ---------------------------------------------------------------------------
__global__ void dftw_kernel(const float* __restrict__ x, const float* __restrict__ FW,
                            float* __restrict__ Xw) {
    __shared__ float As[16 * 512];                       // 32 KB A stage
    const int row0 = blockIdx.x * 16;
    const float* xrow = x + (size_t)row0 * 512;
    for (int idx = threadIdx.x; idx < 16 * 512 / 4; idx += blockDim.x)
        ((float4*)As)[idx] = ((const float4*)xrow)[idx];
    __syncthreads();

    const int wave = threadIdx.x >> 5, lane = threadIdx.x & 31;
    const int n0 = wave * 32;
    v8f acc0 = {}, acc1 = {};
    for (int k = 0; k < 512; k += 4) {
        v2f a  = load_a_tile(As + k, 512, lane);
        v2f b0 = load_b_tile(FW + (size_t)k * 128 + n0,      128, lane);
        v2f b1 = load_b_tile(FW + (size_t)k * 128 + n0 + 16, 128, lane);
        acc0 = wmma_f32(a, b0, acc0);
        acc1 = wmma_f32(a, b1, acc1);
    }
    float* crow = Xw + (size_t)row0 * 128;
    const int n = lane & 15, mb = (lane >> 4) * 8;
    for (int v = 0; v < 8; ++v) {
        crow[(size_t)(mb + v) * 128 + n0 + n]      = acc0[v];
        crow[(size_t)(mb + v) * 128 + n0 + 16 + n] = acc1[v];
    }
}

// ---------------------------------------------------------------------------
// Kernel 2: forward DFT along H (complex), truncated to 64 modes.
//   Xs[c,m1,m2] = sum_h (cos - i sin)(2pi m1 h/512) * Xw[c,h,m2]
//   grid (64 c, 8 m1-chunks), block 256
// ---------------------------------------------------------------------------
__global__ void dfth_kernel(const float* __restrict__ Xw, const float* __restrict__ cosH,
                            const float* __restrict__ sinH, float* __restrict__ Xs) {
    const int c = blockIdx.x, m1base = blockIdx.y * 8;
    const int m2 = threadIdx.x & 63, sub = threadIdx.x >> 6;
    const float* xw = Xw + (size_t)c * 512 * 128 + 2 * m2;
    for (int mm = sub; mm < 8; mm += 4) {
        const int m1 = m1base + mm;
        const float* cp = cosH + (size_t)m1 * 512;
        const float* sp = sinH + (size_t)m1 * 512;
        float accR = 0.0f, accI = 0.0f;
        for (int h = 0; h < 512; ++h) {
            float ct = cp[h], st = sp[h];
            float xr = xw[(size_t)h * 128], xi = xw[(size_t)h * 128 + 1];
            accR += ct * xr + st * xi;
            accI += ct * xi - st * xr;
        }
        Xs[((size_t)c * 64 + m1) * 128 + 2 * m2]     = accR;
        Xs[((size_t)c * 64 + m1) * 128 + 2 * m2 + 1] = accI;
    }
}

// ---------------------------------------------------------------------------
// Kernel 3: per-mode complex channel mix -- THE bandwidth kernel (134 MB of
// weights streamed exactly once). 4-way split over input channels
// (blockIdx.z) for 4x wave concurrency; partials summed in kernel 4.
//   YsP[z][o,m1,m2] = sum_{i in z-chunk} Xs[i,m1,m2] * (wR + i wI)[i,o,m1,m2]
//   grid (64 m1, 4 o-chunks of 16, ISPLIT), block 256; float4 weight loads
// ---------------------------------------------------------------------------
__global__ void modemul_kernel(const float* __restrict__ Xs, const float* __restrict__ wre,
                               const float* __restrict__ wim, float* __restrict__ YsP) {
    const int m1 = blockIdx.x, obase = blockIdx.y * 16;
    const int ibase = blockIdx.z * ICHUNK;
    __shared__ float xsR[ICHUNK * 64], xsI[ICHUNK * 64];   // 8 KB
    for (int idx = threadIdx.x; idx < ICHUNK * 64; idx += blockDim.x) {
        int i = idx >> 6, m2 = idx & 63;
        const float* p = Xs + ((size_t)(ibase + i) * 64 + m1) * 128 + 2 * m2;
        xsR[idx] = p[0];
        xsI[idx] = p[1];
    }
    __syncthreads();

    const int m2q = (threadIdx.x & 15) * 4;
    const int o   = obase + (threadIdx.x >> 4);
    float aR[4] = {0, 0, 0, 0}, aI[4] = {0, 0, 0, 0};
#pragma unroll 4
    for (int i = 0; i < ICHUNK; ++i) {
        size_t base = (((size_t)(ibase + i) * 64 + o) * 64 + m1) * 64 + m2q;
        float4 wr = *(const float4*)(wre + base);
        float4 wi = *(const float4*)(wim + base);
        const float* xr = xsR + i * 64 + m2q;
        const float* xi = xsI + i * 64 + m2q;
        aR[0] += xr[0] * wr.x - xi[0] * wi.x;  aI[0] += xr[0] * wi.x + xi[0] * wr.x;
        aR[1] += xr[1] * wr.y - xi[1] * wi.y;  aI[1] += xr[1] * wi.y + xi[1] * wr.y;
        aR[2] += xr[2] * wr.z - xi[2] * wi.z;  aI[2] += xr[2] * wi.z + xi[2] * wr.z;
        aR[3] += xr[3] * wr.w - xi[3] * wi.w;  aI[3] += xr[3] * wi.w + xi[3] * wr.w;
    }
    float* yp = YsP + (size_t)blockIdx.z * (OUT_C * 64 * 128)
                    + ((size_t)o * 64 + m1) * 128 + 2 * m2q;
    for (int j = 0; j < 4; ++j) { yp[2 * j] = aR[j]; yp[2 * j + 1] = aI[j]; }
}

// ---------------------------------------------------------------------------
// Kernel 4: inverse DFT along H (expand 64 modes -> 512 h, complex),
// summing the ISPLIT partial Ys buffers during the LDS staging pass.
//   Z[o,h,m2] = sum_m1 (cos + i sin)(2pi m1 h/512) * Ys[o,m1,m2]
//   grid (64 o, 64 h-chunks of 8), block 256
// ---------------------------------------------------------------------------
__global__ void idfth_kernel(const float* __restrict__ YsP, const float* __restrict__ cosH,
                             const float* __restrict__ sinH, float* __restrict__ Z) {
    const int o = blockIdx.x, hbase = blockIdx.y * 8;
    __shared__ float ysR[64 * 64], ysI[64 * 64];         // 32 KB
    for (int idx = threadIdx.x; idx < 64 * 64; idx += blockDim.x) {
        int m1 = idx >> 6, m2 = idx & 63;
        size_t off = ((size_t)o * 64 + m1) * 128 + 2 * m2;
        float r = 0.0f, im = 0.0f;
        for (int z = 0; z < ISPLIT; ++z) {
            const float* p = YsP + (size_t)z * (OUT_C * 64 * 128) + off;
            r  += p[0];
            im += p[1];
        }
        ysR[idx] = r;
        ysI[idx] = im;
    }
    __syncthreads();

    const int m2 = threadIdx.x & 63, sub = threadIdx.x >> 6;
    for (int hh = sub; hh < 8; hh += 4) {
        const int h = hbase + hh;
        float accR = 0.0f, accI = 0.0f;
        for (int m1 = 0; m1 < 64; ++m1) {
            float ct = cosH[(size_t)m1 * 512 + h], st = sinH[(size_t)m1 * 512 + h];
            float yr = ysR[m1 * 64 + m2], yi = ysI[m1 * 64 + m2];
            accR += ct * yr - st * yi;
            accI += ct * yi + st * yr;
        }
        Z[((size_t)o * 512 + h) * 128 + 2 * m2]     = accR;
        Z[((size_t)o * 512 + h) * 128 + 2 * m2 + 1] = accI;
    }
}

// ---------------------------------------------------------------------------
// Kernel 5 (WMMA): fused real iDFT along W + skip GEMM + biases + GELU.
//   out[o,h,w] = gelu( sum_{k<128} Zcat[o,h,k]*DW[k,w]
//                    + sum_{i<64}  w_skip[o,i]*x[i,h,w] + b_conv[o]+b_skip[o] )
//   grid 512 (one h per block), block 256 (8 waves x 64-wide w chunks)
// ---------------------------------------------------------------------------
__global__ void invw_skip_gelu_kernel(const float* __restrict__ Z, const float* __restrict__ DW,
                                      const float* __restrict__ x, const float* __restrict__ wskip,
                                      const float* __restrict__ bconv, const float* __restrict__ bskip,
                                      float* __restrict__ out) {
    const int h = blockIdx.x;
    __shared__ float Az[64 * 128];                       // 32 KB spectral A
    __shared__ float Ak[64 * 64];                        // 16 KB skip A
    for (int idx = threadIdx.x; idx < 64 * 128 / 4; idx += blockDim.x) {
        int o = idx >> 5, kq = idx & 31;
        ((float4*)(Az + o * 128))[kq] =
            ((const float4*)(Z + ((size_t)o * 512 + h) * 128))[kq];
    }
    for (int idx = threadIdx.x; idx < 64 * 64 / 4; idx += blockDim.x)
        ((float4*)Ak)[idx] = ((const float4*)wskip)[idx];
    __syncthreads();

    const int wave = threadIdx.x >> 5, lane = threadIdx.x & 31;
    const int n0 = wave * 64;                            // this wave's w chunk
    const float* xh = x + (size_t)h * 512;

    for (int mt = 0; mt < 4; ++mt) {
        const int m0 = mt * 16;
        v8f acc[4] = {{}, {}, {}, {}};
        // spectral contraction: K = 128 (interleaved R/I)
        for (int k = 0; k < 128; k += 4) {
            v2f a = load_a_tile(Az + m0 * 128 + k, 128, lane);
            for (int j = 0; j < 4; ++j) {
                v2f b = load_b_tile(DW + (size_t)k * 512 + n0 + 16 * j, 512, lane);
                acc[j] = wmma_f32(a, b, acc[j]);
            }
        }
        // skip contraction: K = 64, B = x[:, h, :] with row stride HW
        for (int k = 0; k < 64; k += 4) {
            v2f a = load_a_tile(Ak + m0 * 64 + k, 64, lane);
            for (int j = 0; j < 4; ++j) {
                v2f b = load_b_tile(xh + (size_t)k * HW + n0 + 16 * j, (size_t)HW, lane);
                acc[j] = wmma_f32(a, b, acc[j]);
            }
        }
        // epilogue: biases + tanh-gelu, store
        const int n = lane & 15, mb = m0 + (lane >> 4) * 8;
        for (int j = 0; j < 4; ++j) {
            for (int v = 0; v < 8; ++v) {
                int m = mb + v, w = n0 + 16 * j + n;
                float val = acc[j][v] + bconv[m] + bskip[m];
                float c3  = val * val * val;
                float g   = 0.5f * val *
                            (1.0f + tanhf(0.7978845608028654f * (val + 0.044715f * c3)));
                out[(size_t)m * HW + (size_t)h * 512 + w] = g;
            }
        }
    }
}

// ---------------------------------------------------------------------------
// Host orchestration
// ---------------------------------------------------------------------------
extern "C" void kernel_launch(void* const* d_in, const int* in_sizes, int n_in,
                              void* d_out, int out_size, void* d_ws, size_t ws_size,
                              hipStream_t stream) {
    const float* x     = (const float*)d_in[0];   // (64,512,512)
    const float* w_re  = (const float*)d_in[1];   // (64,64,64,64)
    const float* w_im  = (const float*)d_in[2];   // (64,64,64,64)
    const float* b_cv  = (const float*)d_in[3];   // (64)
    const float* w_sk  = (const float*)d_in[4];   // (64,64)
    const float* b_sk  = (const float*)d_in[5];   // (64)
    float* out = (float*)d_out;

    float* ws   = (float*)d_ws;
    float* FW   = ws;                         // 512*128          = 65536
    float* cosH = FW   + 65536;               // 64*512           = 32768
    float* sinH = cosH + 32768;               // 64*512           = 32768
    float* DW   = sinH + 32768;               // 128*512          = 65536
    float* Xw   = DW   + 65536;               // 32768*128        = 4194304
    float* Xs   = Xw   + 4194304;             // 64*64*128        = 524288
    float* YsP  = Xs   + 524288;              // ISPLIT*64*64*128 = 2097152
    float* Z    = YsP  + 2097152;             // 64*512*128       = 4194304

    twiddle_kernel<<<128, 256, 0, stream>>>(FW, cosH, sinH, DW);
    dftw_kernel<<<2048, 128, 0, stream>>>(x, FW, Xw);
    dfth_kernel<<<dim3(64, 8), 256, 0, stream>>>(Xw, cosH, sinH, Xs);
    modemul_kernel<<<dim3(64, 4, ISPLIT), 256, 0, stream>>>(Xs, w_re, w_im, YsP);
    idfth_kernel<<<dim3(64, 64), 256, 0, stream>>>(YsP, cosH, sinH, Z);
    invw_skip_gelu_kernel<<<512, 256, 0, stream>>>(Z, DW, x, w_sk, b_cv, b_sk, out);
}